// AELoss_18227841204943
// MI455X (gfx1250) — compile-verified
//
#include <hip/hip_runtime.h>

#define B_ 64
#define P_ 30
#define K_ 17
#define H_ 256
#define W_ 256
#define PK_ (P_ * K_)   // 510

typedef __attribute__((ext_vector_type(2))) float v2f;
typedef __attribute__((ext_vector_type(8))) float v8f;

// One workgroup per batch image. 512 threads = 16 wave32s.
__global__ __launch_bounds__(512) void ae_loss_per_image(
    const float* __restrict__ tags,     // [B,K,H,W]
    const int*   __restrict__ joints,   // [B,P,K,2]
    const int*   __restrict__ jvalid,   // [B,P,K]
    const int*   __restrict__ pvalid,   // [B,P]
    float*       __restrict__ ws)       // [B,2] (push, pull)
{
    __shared__ float sVM[PK_];      // v * m per (p,k)
    __shared__ float sM[PK_];       // m per (p,k)
    __shared__ float sMean[32];     // per-person mean (0 if invalid), padded to 32
    __shared__ float sPM[32];       // person-valid mask, padded
    __shared__ float sPullP[32];    // per-person pull * pm
    __shared__ float sPushPart[128];
    __shared__ float sN, sPull;

    const int b = blockIdx.x;
    const int t = threadIdx.x;

    // ---- Phase 1: sparse gather of tag values at joint locations ----
    if (t < PK_) {
        const int p = t / K_;
        const int k = t - p * K_;
        const int base = (b * P_ + p) * K_ + k;
        const int jv = jvalid[base];
        const int pv = pvalid[b * P_ + p];
        const float m = (jv > 0 && pv > 0) ? 1.0f : 0.0f;
        float v = 0.0f;
        if (m > 0.0f) {
            const int x = joints[base * 2 + 0];
            const int y = joints[base * 2 + 1];
            v = tags[((((long)b) * K_ + k) * H_ + x) * W_ + y];
        }
        sVM[t] = v;   // equals v*m
        sM[t]  = m;
    }
    if (t < 32) { sMean[t] = 0.0f; sPM[t] = 0.0f; sPullP[t] = 0.0f; }
    __syncthreads();

    // ---- Phase 2: per-person mean + pull (30 threads, fixed-order sums) ----
    if (t < P_) {
        float sv = 0.0f, cnt = 0.0f;
        #pragma unroll
        for (int k = 0; k < K_; ++k) { sv += sVM[t * K_ + k]; cnt += sM[t * K_ + k]; }
        const float safe = fmaxf(cnt, 1.0f);
        const float mean = sv / safe;
        float pp = 0.0f;
        #pragma unroll
        for (int k = 0; k < K_; ++k) {
            const float d = sVM[t * K_ + k] - mean;
            pp += sM[t * K_ + k] * d * d;
        }
        pp /= safe;
        const float pm = (cnt > 0.0f) ? 1.0f : 0.0f;
        sMean[t]  = pm * mean;
        sPM[t]    = pm;
        sPullP[t] = pm * pp;
    }
    __syncthreads();

    if (t == 0) {
        float n = 0.0f, ps = 0.0f;
        for (int p = 0; p < P_; ++p) { n += sPM[p]; ps += sPullP[p]; }
        sN = n;
        sPull = (n > 0.0f) ? ps / fmaxf(n, 1.0f) : 0.0f;
    }
    __syncthreads();

    // ---- Phase 3: 32x32 pairwise diff matrix via V_WMMA_F32_16X16X4_F32 ----
    // D[i][j] = mean_i - mean_j from A = [mean_i, -1, 0, 0] (16x4),
    //                               B = [[1..1],[mean_j],[0],[0]] (4x16).
    // Waves 0..3 each own one 16x16 tile; EXEC all-1s inside each wave.
    const int wave = t >> 5;
    const int lane = t & 31;
    if (wave < 4) {
        const int tr = wave >> 1;   // tile row (i block)
        const int tc = wave & 1;    // tile col (j block)
        v2f A, Bf;
        if (lane < 16) {
            // A layout: vgpr0 = K=0 (lanes 0-15), vgpr1 = K=1 (lanes 0-15)
            A.x = sMean[tr * 16 + lane];
            A.y = -1.0f;
            // B layout: vgpr0 = row K=0 (lanes 0-15), vgpr1 = row K=1
            Bf.x = 1.0f;
            Bf.y = sMean[tc * 16 + lane];
        } else {
            // lanes 16-31 hold K=2,3 slices -> all zero
            A.x = 0.0f; A.y = 0.0f; Bf.x = 0.0f; Bf.y = 0.0f;
        }
        v8f C = {};
        C = __builtin_amdgcn_wmma_f32_16x16x4_f32(
                /*neg_a=*/false, A, /*neg_b=*/false, Bf,
                /*c_mod=*/(short)0, C, /*reuse_a=*/false, /*reuse_b=*/false);

        // C/D layout: VGPR r -> (M=r, N=lane) for lanes 0-15, (M=r+8, N=lane-16) otherwise
        const int jn    = tc * 16 + (lane & 15);
        const int ibase = tr * 16 + ((lane >> 4) << 3);
        const float pmj = sPM[jn];
        float acc = 0.0f;
        #pragma unroll
        for (int r = 0; r < 8; ++r) {
            const float d = C[r];
            acc += sPM[ibase + r] * pmj * expf(-d * d);
        }
        sPushPart[t] = acc;
    }
    __syncthreads();

    if (t == 0) {
        float s = 0.0f;
        for (int i = 0; i < 128; ++i) s += sPushPart[i];   // fixed order: deterministic
        const float n = sN;
        s -= n;                                            // remove diagonal exp(0) terms
        const float push = (n > 1.0f) ? s / fmaxf((n - 1.0f) * n, 1.0f) * 0.5f : 0.0f;
        ws[b * 2 + 0] = push;
        ws[b * 2 + 1] = sPull;
    }
}

__global__ __launch_bounds__(64) void ae_loss_reduce(
    const float* __restrict__ ws, float* __restrict__ out)
{
    __shared__ float sp[B_], sl[B_];
    const int t = threadIdx.x;
    sp[t] = ws[t * 2 + 0];
    sl[t] = ws[t * 2 + 1];
    __syncthreads();
    if (t == 0) {
        float a = 0.0f, c = 0.0f;
        for (int i = 0; i < B_; ++i) { a += sp[i]; c += sl[i]; }
        out[0] = a * (1.0f / B_);   // push mean
        out[1] = c * (1.0f / B_);   // pull mean
    }
}

extern "C" void kernel_launch(void* const* d_in, const int* in_sizes, int n_in,
                              void* d_out, int out_size, void* d_ws, size_t ws_size,
                              hipStream_t stream) {
    const float* tags   = (const float*)d_in[0];
    const int*   joints = (const int*)d_in[1];
    const int*   jval   = (const int*)d_in[2];
    const int*   pval   = (const int*)d_in[3];
    float* ws = (float*)d_ws;   // needs B_*2 floats = 512 bytes

    ae_loss_per_image<<<B_, 512, 0, stream>>>(tags, joints, jval, pval, ws);
    ae_loss_reduce<<<1, 64, 0, stream>>>(ws, (float*)d_out);
}